// QuantumOperation_67894843015387
// MI455X (gfx1250) — compile-verified
//
#include <hip/hip_runtime.h>

// D = A(16x4) * B(4x16) + C  via v_wmma_f32_16x16x4_f32 (wave32, gfx1250)
typedef __attribute__((ext_vector_type(2))) float v2f;
typedef __attribute__((ext_vector_type(8))) float v8f;

#define BIT_Q12  (1u << 13)        // element stride of qubit-12 axis
#define BIT_Q5   (1u << 20)        // element stride of qubit-5 axis
#define REST_N   (1u << 24)        // 2^22 non-support qubit combos * batch 4
#define GROUPS   (REST_N / 16u)    // 16 columns per WMMA -> 2^20 groups
#define GROUPS_PER_WAVE 8u
#define WAVES_PER_BLOCK 8u
#define THREADS (WAVES_PER_BLOCK * 32u)

// Insert zero bits at address positions 13 (qubit 12) and 20 (qubit 5)
// of the 24-bit rest index.
__device__ __forceinline__ unsigned expand_rest(unsigned r) {
    unsigned low  = r & 0x1FFFu;        // -> addr bits 0..12
    unsigned mid  = (r >> 13) & 0x3Fu;  // -> addr bits 14..19
    unsigned high = r >> 19;            // -> addr bits 21..25
    return low | (mid << 14) | (high << 21);
}

__global__ __launch_bounds__(THREADS)
void apply_2q_gate_wmma(const float* __restrict__ state,
                        const float* __restrict__ gate,
                        float* __restrict__ out) {
    const unsigned tid  = blockIdx.x * blockDim.x + threadIdx.x;
    const unsigned lane = threadIdx.x & 31u;
    const unsigned wave = tid >> 5;

    const unsigned m  = lane & 15u;        // A: row M / B,D: column N
    const unsigned kb = (lane >> 4) << 1;  // K base for this lane half: 0 or 2

    // A matrix: gate rows 0..3, rows 4..15 zero.
    // lane<16: V0=A[m][0], V1=A[m][1]; lane>=16: V0=A[m][2], V1=A[m][3]
    v2f a;
    a.x = (m < 4u) ? gate[m * 4u + kb]      : 0.0f;
    a.y = (m < 4u) ? gate[m * 4u + kb + 1u] : 0.0f;

    // K = 2*b5 + b12  ->  element offset: b5*BIT_Q5 + b12*BIT_Q12.
    // This lane supplies K=kb (V0) and K=kb+1 (V1): kb selects the b5 bit,
    // +1 adds the b12 bit.
    const unsigned koff = (kb & 2u) ? BIT_Q5 : 0u;

    const unsigned g0 = wave * GROUPS_PER_WAVE;
    for (unsigned i = 0; i < GROUPS_PER_WAVE; ++i) {
        const unsigned r    = (g0 + i) * 16u + m;   // rest index of column N=m
        const unsigned base = expand_rest(r);

        // B matrix 4x16: lanes 0-15 V0=K0,V1=K1; lanes 16-31 V0=K2,V1=K3
        v2f b;
        b.x = __builtin_nontemporal_load(state + base + koff);
        b.y = __builtin_nontemporal_load(state + base + koff + BIT_Q12);

        v8f c = {};
        c = __builtin_amdgcn_wmma_f32_16x16x4_f32(
                /*neg_a=*/false, a, /*neg_b=*/false, b,
                /*c_mod=*/(short)0, c, /*reuse_a=*/false, /*reuse_b=*/false);

        // D: VGPR r, lanes 0-15 => (M=r, N=lane). Rows 0..3 are the outputs;
        // row r goes to support offset {0, +Q12, +Q5, +Q5+Q12}.
        if (lane < 16u) {
            __builtin_nontemporal_store(c[0], out + base);
            __builtin_nontemporal_store(c[1], out + base + BIT_Q12);
            __builtin_nontemporal_store(c[2], out + base + BIT_Q5);
            __builtin_nontemporal_store(c[3], out + base + BIT_Q5 + BIT_Q12);
        }
    }
}

extern "C" void kernel_launch(void* const* d_in, const int* in_sizes, int n_in,
                              void* d_out, int out_size, void* d_ws, size_t ws_size,
                              hipStream_t stream) {
    (void)in_sizes; (void)n_in; (void)out_size; (void)d_ws; (void)ws_size;
    const float* state = (const float*)d_in[0];  // 2^26 floats
    const float* gate  = (const float*)d_in[1];  // 16 floats
    float*       out   = (float*)d_out;          // 2^26 floats

    const unsigned blocks = GROUPS / (WAVES_PER_BLOCK * GROUPS_PER_WAVE); // 16384
    apply_2q_gate_wmma<<<blocks, THREADS, 0, stream>>>(state, gate, out);
}